// BahdanauAttention_85925115724085
// MI455X (gfx1250) — compile-verified
//
#include <hip/hip_runtime.h>
#include <hip/hip_bf16.h>
#include <math.h>

#define BATCH 64
#define KLEN  2048
#define HID   1024
#define VDIM  1024

typedef float v2f __attribute__((ext_vector_type(2)));
typedef float v8f __attribute__((ext_vector_type(8)));

// -------- Kernel 1: q_proj[b,h] = sum_d query[b,d] * Wq[h,d] via f32 WMMA ----
// GEMM M=64 (batch) x N=1024 (h) x K=1024 (d). One wave = one 16x16 tile,
// K-loop of 256 x V_WMMA_F32_16X16X4_F32.
// 32-bit A (16x4) layout: lanes 0-15 hold M=lane, VGPR0/1 = K0/K1;
//                         lanes 16-31 hold M=lane-16, K2/K3. B mirrored over N.
__global__ __launch_bounds__(32) void qproj_wmma(const float* __restrict__ query,
                                                 const float* __restrict__ Wq,
                                                 float* __restrict__ qproj) {
    const int lane = threadIdx.x;
    const int tN = blockIdx.x * 16;   // h tile
    const int tM = blockIdx.y * 16;   // batch tile
    const int l  = lane & 15;
    const int kh = (lane >> 4) << 1;  // 0 or 2: which K pair this half-wave holds

    const float* arow = query + (size_t)(tM + l) * HID + kh;  // A[m, k]
    const float* brow = Wq    + (size_t)(tN + l) * HID + kh;  // B[k, n] = Wq[n, k]

    v8f c = {};
    #pragma unroll 4
    for (int k = 0; k < HID; k += 4) {
        v2f a = *(const v2f*)(arow + k);
        v2f b = *(const v2f*)(brow + k);
        // D = A(16x4,f32) x B(4x16,f32) + C
        c = __builtin_amdgcn_wmma_f32_16x16x4_f32(false, a, false, b,
                                                  (short)0, c, false, false);
    }
    // C/D layout: VGPR j -> M = j (lanes 0-15) or 8+j (lanes 16-31); N = lane&15
    const int row = tM + ((lane >> 4) << 3);
    const int col = tN + l;
    #pragma unroll
    for (int j = 0; j < 8; ++j)
        qproj[(size_t)(row + j) * HID + col] = c[j];
}

// -------- fast tanh: stable for all x, lowers to v_exp_f32 / v_rcp_f32 -------
__device__ __forceinline__ float tanh_fast(float x) {
    float a = fabsf(x);
    float e = __expf(-2.0f * a);          // in (0, 1]
    float t = (1.0f - e) / (1.0f + e);    // tanh(|x|)
    return copysignf(t, x);
}

// -------- Kernel 2: energy[b,k] = sum_h tanh(qproj[b,h]+key[b,k,h]) * We[h] --
// grid = (KLEN/32, BATCH); block = 256 (8 waves). Wave handles 4 k-rows,
// lanes cooperate across h with float4 coalesced loads (4KB/row).
__global__ __launch_bounds__(256) void energy_kernel(const float* __restrict__ key,
                                                     const float* __restrict__ qproj,
                                                     const float* __restrict__ We,
                                                     float* __restrict__ energy) {
    __shared__ float s_qp[HID];
    __shared__ float s_we[HID];
    const int b   = blockIdx.y;
    const int tid = threadIdx.x;
    for (int i = tid; i < HID; i += 256) {
        s_qp[i] = qproj[(size_t)b * HID + i];
        s_we[i] = We[i];
    }
    __syncthreads();

    const int wave = tid >> 5;
    const int lane = tid & 31;
    const int k0   = blockIdx.x * 32 + wave * 4;
    const float4* s_qp4 = (const float4*)s_qp;
    const float4* s_we4 = (const float4*)s_we;

    for (int i = 0; i < 4; ++i) {
        const int k = k0 + i;
        const float4* krow = (const float4*)(key + ((size_t)b * KLEN + k) * HID);
        float sum = 0.0f;
        #pragma unroll
        for (int j = 0; j < 8; ++j) {
            const int h4 = lane + j * 32;      // float4 index, coalesced per wave
            float4 kv = krow[h4];
            float4 qp = s_qp4[h4];
            float4 we = s_we4[h4];
            sum += we.x * tanh_fast(qp.x + kv.x);
            sum += we.y * tanh_fast(qp.y + kv.y);
            sum += we.z * tanh_fast(qp.z + kv.z);
            sum += we.w * tanh_fast(qp.w + kv.w);
        }
        #pragma unroll
        for (int off = 16; off > 0; off >>= 1)
            sum += __shfl_xor(sum, off, 32);
        if (lane == 0) energy[(size_t)b * KLEN + k] = sum;
    }
}

// -------- Kernel 3: masked softmax over k per batch; writes attn to d_out ----
__global__ __launch_bounds__(256) void softmax_kernel(const float* __restrict__ energy,
                                                      const int* __restrict__ mask,
                                                      float* __restrict__ attn) {
    __shared__ float s_red[8];
    __shared__ float s_max, s_sum;
    const int b   = blockIdx.x;
    const int tid = threadIdx.x;
    const int wave = tid >> 5, lane = tid & 31;

    float e[8];
    float mx = -INFINITY;
    #pragma unroll
    for (int i = 0; i < 8; ++i) {
        const int k = tid + i * 256;
        float v = energy[(size_t)b * KLEN + k];
        if (mask[(size_t)b * KLEN + k] == 0) v = -INFINITY;
        e[i] = v;
        mx = fmaxf(mx, v);
    }
    #pragma unroll
    for (int off = 16; off > 0; off >>= 1)
        mx = fmaxf(mx, __shfl_xor(mx, off, 32));
    if (lane == 0) s_red[wave] = mx;
    __syncthreads();
    if (tid == 0) {
        float m = s_red[0];
        #pragma unroll
        for (int w = 1; w < 8; ++w) m = fmaxf(m, s_red[w]);
        s_max = m;
    }
    __syncthreads();
    mx = s_max;

    float sum = 0.0f;
    #pragma unroll
    for (int i = 0; i < 8; ++i) {
        e[i] = __expf(e[i] - mx);
        sum += e[i];
    }
    #pragma unroll
    for (int off = 16; off > 0; off >>= 1)
        sum += __shfl_xor(sum, off, 32);
    __syncthreads();               // s_red safe to reuse
    if (lane == 0) s_red[wave] = sum;
    __syncthreads();
    if (tid == 0) {
        float t = 0.0f;
        #pragma unroll
        for (int w = 0; w < 8; ++w) t += s_red[w];
        s_sum = t;
    }
    __syncthreads();
    const float inv = 1.0f / s_sum;
    #pragma unroll
    for (int i = 0; i < 8; ++i)
        attn[(size_t)b * KLEN + tid + i * 256] = e[i] * inv;
}

// -------- Kernel 4: partial context over a 256-wide k-chunk ------------------
// grid = (8 chunks, BATCH); block = 256; thread owns 4 consecutive v (float4).
__global__ __launch_bounds__(256) void context_partial(const float* __restrict__ value,
                                                       const float* __restrict__ attn,
                                                       float* __restrict__ part) {
    const int c   = blockIdx.x;
    const int b   = blockIdx.y;
    const int tid = threadIdx.x;
    const int v4  = tid;  // float4 index: v = 4*tid

    const float*  arow  = attn + (size_t)b * KLEN + c * 256;
    const float4* vbase = (const float4*)(value + ((size_t)b * KLEN + c * 256) * VDIM) + v4;

    float4 acc = make_float4(0.f, 0.f, 0.f, 0.f);
    #pragma unroll 4
    for (int k = 0; k < 256; ++k) {
        const float  w   = arow[k];
        const float4 val = vbase[(size_t)k * (VDIM / 4)];
        acc.x += w * val.x;
        acc.y += w * val.y;
        acc.z += w * val.z;
        acc.w += w * val.w;
    }
    ((float4*)part)[((size_t)c * BATCH + b) * (VDIM / 4) + v4] = acc;
}

// -------- Kernel 5: fixed-order reduction of the 8 partials (deterministic) --
__global__ __launch_bounds__(256) void context_reduce(const float* __restrict__ part,
                                                      float* __restrict__ out) {
    const int idx = blockIdx.x * 256 + threadIdx.x;  // over BATCH*VDIM = 65536
    float s = 0.0f;
    #pragma unroll
    for (int c = 0; c < 8; ++c)
        s += part[(size_t)c * (BATCH * VDIM) + idx];
    out[idx] = s;
}

extern "C" void kernel_launch(void* const* d_in, const int* in_sizes, int n_in,
                              void* d_out, int out_size, void* d_ws, size_t ws_size,
                              hipStream_t stream) {
    const float* query = (const float*)d_in[0];   // (64,1,1024)
    const float* key   = (const float*)d_in[1];   // (64,2048,1024)
    const float* value = (const float*)d_in[2];   // (64,2048,1024)
    const int*   mask  = (const int*)  d_in[3];   // (64,1,2048)
    const float* Wq    = (const float*)d_in[4];   // (1024,1024)
    const float* We    = (const float*)d_in[5];   // (1024,)

    float* ws       = (float*)d_ws;
    float* qproj    = ws;                                   // 65536 f32
    float* energy   = ws + (size_t)BATCH * HID;             // 131072 f32
    float* ctx_part = energy + (size_t)BATCH * KLEN;        // 8*65536 f32

    float* out_ctx  = (float*)d_out;                        // context: 64*1024
    float* out_attn = out_ctx + (size_t)BATCH * VDIM;       // attn:    64*2048

    qproj_wmma     <<<dim3(HID / 16, BATCH / 16), 32,  0, stream>>>(query, Wq, qproj);
    energy_kernel  <<<dim3(KLEN / 32, BATCH),     256, 0, stream>>>(key, qproj, We, energy);
    softmax_kernel <<<BATCH,                      256, 0, stream>>>(energy, mask, out_attn);
    context_partial<<<dim3(KLEN / 256, BATCH),    256, 0, stream>>>(value, out_attn, ctx_part);
    context_reduce <<<(BATCH * VDIM) / 256,       256, 0, stream>>>(ctx_part, out_ctx);
}